// MultiModalFusionGAT_75179107549411
// MI455X (gfx1250) — compile-verified
//
#include <hip/hip_runtime.h>
#include <hip/hip_bf16.h>
#include <math.h>

#define H 8
#define DH 16
#define HID 128

typedef float v2f __attribute__((ext_vector_type(2)));
typedef float v8f __attribute__((ext_vector_type(8)));

// ---------------------------------------------------------------------------
// Packed weight layout for WMMA B-operand:
//   Wp[(k/2)*128 + n] = { W[k,n], W[k+1,n] }   (v2f elements)
// Lane (k-step k0, koff, col n) loads ONE coalesced 8-byte element.
// ---------------------------------------------------------------------------

// Fuse per-head relation matrices into projection weights AND emit packed:
//   kr = x @ (Wk[:,hblk] @ a_rel[h]) + (bk[hblk] @ a_rel[h])
__global__ void fuse_weights_kernel(const float* __restrict__ Wk, const float* __restrict__ bk,
                                    const float* __restrict__ Wv, const float* __restrict__ bv,
                                    const float* __restrict__ a_rel, const float* __restrict__ m_rel,
                                    float* __restrict__ Wka_p, float* __restrict__ bka,
                                    float* __restrict__ Wva_p, float* __restrict__ bva) {
    int tid = blockIdx.x * blockDim.x + threadIdx.x;
    if (tid >= 4 * HID * HID) return;
    int t   = tid >> 14;          // node type 0..3
    int rem = tid & 16383;
    int c   = rem >> 7;           // input dim (k)
    int n   = rem & 127;          // output col
    int h   = n >> 4;
    int e   = n & 15;
    // node type -> relation where it is the src: user->2, item->3, taste->0, image->1
    const int relmap[4] = {2, 3, 0, 1};
    int r = relmap[t];
    const float* A  = a_rel + ((size_t)r * H + h) * DH * DH;   // [16,16] (d,e)
    const float* M  = m_rel + ((size_t)r * H + h) * DH * DH;
    const float* wk = Wk + (size_t)t * HID * HID + (size_t)c * HID + h * DH;
    const float* wv = Wv + (size_t)t * HID * HID + (size_t)c * HID + h * DH;
    float sk = 0.f, sv = 0.f;
#pragma unroll
    for (int d = 0; d < DH; ++d) {
        sk += wk[d] * A[d * DH + e];
        sv += wv[d] * M[d * DH + e];
    }
    size_t po = (size_t)t * HID * HID + (((size_t)(c >> 1) * HID + n) << 1) + (c & 1);
    Wka_p[po] = sk;
    Wva_p[po] = sv;
    if (c == 0) {
        const float* bkp = bk + t * HID + h * DH;
        const float* bvp = bv + t * HID + h * DH;
        float sbk = 0.f, sbv = 0.f;
#pragma unroll
        for (int d = 0; d < DH; ++d) {
            sbk += bkp[d] * A[d * DH + e];
            sbv += bvp[d] * M[d * DH + e];
        }
        bka[t * HID + n] = sbk;
        bva[t * HID + n] = sbv;
    }
}

// Repack plain [nmat,128,128] weights into K-pair-interleaved layout.
__global__ void repack_kernel(const float* __restrict__ in, float* __restrict__ out, int nmat) {
    int tid = blockIdx.x * blockDim.x + threadIdx.x;
    if (tid >= nmat * HID * HID) return;
    int m = tid >> 14;
    int rem = tid & 16383;
    int k = rem >> 7;
    int n = rem & 127;
    out[(size_t)m * HID * HID + (((size_t)(k >> 1) * HID + n) << 1) + (k & 1)] = in[tid];
}

// ---------------------------------------------------------------------------
// WMMA fp32 GEMM: Out[N,128] = X[N,128] @ W[128,128] + bias, packed W.
// One wave32 per block computes a 32x128 tile (two A fragments share B).
// ---------------------------------------------------------------------------
__global__ __launch_bounds__(32) void gemm32_kernel(const float* __restrict__ X,
                                                    const float* __restrict__ Wp,
                                                    const float* __restrict__ bias,
                                                    float* __restrict__ Out, int N) {
    __shared__ float xs[32 * HID];  // 16 KB
    const int lane = threadIdx.x;
    const int row0 = blockIdx.x * 32;

    // stage 32x128 X tile into LDS with float4s
    const float4* Xv = (const float4*)X;
    float4* xsv = (float4*)xs;
    for (int i = lane; i < 32 * 32; i += 32) {
        int r = i >> 5, c = i & 31;
        int gr = row0 + r;
        float4 v = make_float4(0.f, 0.f, 0.f, 0.f);
        if (gr < N) v = Xv[(size_t)gr * 32 + c];
        xsv[i] = v;
    }
    __syncthreads();

    const int mrow = lane & 15;           // A row in tile
    const int koff = (lane >> 4) * 2;     // 16x4 A layout: lanes 0-15 K=0..1, 16-31 K=2..3
    const int ncol = lane & 15;           // B/C column
    const int mb   = (lane >> 4) * 8;     // C: VGPR j -> M=j (low half) / M=8+j
    const v2f* Wpv = (const v2f*)Wp;
    const bool full = (row0 + 32 <= N);

    for (int nt = 0; nt < 8; ++nt) {
        const int n0 = nt * 16;
        v8f acc0 = {}, acc1 = {};
#pragma unroll 8
        for (int k0 = 0; k0 < HID; k0 += 4) {
            v2f a0 = *(const v2f*)(xs + mrow * HID + k0 + koff);
            v2f a1 = *(const v2f*)(xs + (16 + mrow) * HID + k0 + koff);
            v2f b  = Wpv[(size_t)((k0 + koff) >> 1) * HID + n0 + ncol];
            acc0 = __builtin_amdgcn_wmma_f32_16x16x4_f32(false, a0, false, b, (short)0, acc0,
                                                         false, false);
            acc1 = __builtin_amdgcn_wmma_f32_16x16x4_f32(false, a1, false, b, (short)0, acc1,
                                                         false, false);
        }
        const float bb = bias[n0 + ncol];
        float* o0 = Out + (size_t)(row0 + mb) * HID + n0 + ncol;
        if (full) {
#pragma unroll
            for (int j = 0; j < 8; ++j) {
                o0[(size_t)j * HID] = acc0[j] + bb;
                o0[(size_t)(j + 16) * HID] = acc1[j] + bb;
            }
        } else {
#pragma unroll
            for (int j = 0; j < 8; ++j) {
                int r0 = row0 + mb + j;
                if (r0 < N) Out[(size_t)r0 * HID + n0 + ncol] = acc0[j] + bb;
                int r1 = r0 + 16;
                if (r1 < N) Out[(size_t)r1 * HID + n0 + ncol] = acc1[j] + bb;
            }
        }
    }
}

// ---------------------------------------------------------------------------
// Output GEMM: Out = g * (gelu(A) @ W + bias) + (1-g) * Xres, g = sigmoid(skip)
// ---------------------------------------------------------------------------
__global__ __launch_bounds__(32) void out_gemm_kernel(const float* __restrict__ A,
                                                      const float* __restrict__ Wp,
                                                      const float* __restrict__ bias,
                                                      const float* __restrict__ Xres,
                                                      const float* __restrict__ skipp, int sidx,
                                                      float* __restrict__ Out, int N) {
    __shared__ float xs[32 * HID];
    const int lane = threadIdx.x;
    const int row0 = blockIdx.x * 32;

    // stage gelu(A) tile into LDS (exact gelu: 0.5x(1+erf(x/sqrt2)))
    for (int i = lane; i < 32 * HID; i += 32) {
        int r = i >> 7, c = i & 127;
        int gr = row0 + r;
        float v = 0.f;
        if (gr < N) v = A[(size_t)gr * HID + c];
        xs[i] = 0.5f * v * (1.0f + erff(v * 0.70710678118654752f));
    }
    __syncthreads();

    const float g = 1.0f / (1.0f + __expf(-skipp[sidx]));
    const float og = 1.0f - g;
    const int mrow = lane & 15;
    const int koff = (lane >> 4) * 2;
    const int ncol = lane & 15;
    const int mb   = (lane >> 4) * 8;
    const v2f* Wpv = (const v2f*)Wp;
    const bool full = (row0 + 32 <= N);

    for (int nt = 0; nt < 8; ++nt) {
        const int n0 = nt * 16;
        v8f acc0 = {}, acc1 = {};
#pragma unroll 8
        for (int k0 = 0; k0 < HID; k0 += 4) {
            v2f a0 = *(const v2f*)(xs + mrow * HID + k0 + koff);
            v2f a1 = *(const v2f*)(xs + (16 + mrow) * HID + k0 + koff);
            v2f b  = Wpv[(size_t)((k0 + koff) >> 1) * HID + n0 + ncol];
            acc0 = __builtin_amdgcn_wmma_f32_16x16x4_f32(false, a0, false, b, (short)0, acc0,
                                                         false, false);
            acc1 = __builtin_amdgcn_wmma_f32_16x16x4_f32(false, a1, false, b, (short)0, acc1,
                                                         false, false);
        }
        const float bb = bias[n0 + ncol];
        if (full) {
            const float* xr = Xres + (size_t)(row0 + mb) * HID + n0 + ncol;
            float* o0 = Out + (size_t)(row0 + mb) * HID + n0 + ncol;
#pragma unroll
            for (int j = 0; j < 8; ++j) {
                o0[(size_t)j * HID] = g * (acc0[j] + bb) + og * xr[(size_t)j * HID];
                o0[(size_t)(j + 16) * HID] =
                    g * (acc1[j] + bb) + og * xr[(size_t)(j + 16) * HID];
            }
        } else {
#pragma unroll
            for (int j = 0; j < 8; ++j) {
                int r0 = row0 + mb + j;
                if (r0 < N)
                    Out[(size_t)r0 * HID + n0 + ncol] =
                        g * (acc0[j] + bb) + og * Xres[(size_t)r0 * HID + n0 + ncol];
                int r1 = r0 + 16;
                if (r1 < N)
                    Out[(size_t)r1 * HID + n0 + ncol] =
                        g * (acc1[j] + bb) + og * Xres[(size_t)r1 * HID + n0 + ncol];
            }
        }
    }
}

// ---------------------------------------------------------------------------
// Workspace init (harness poisons d_ws; re-zero accumulators every call)
// ---------------------------------------------------------------------------
__global__ void init_ws_kernel(float* agg_i, int ni, float* agg_u, int nu,
                               float* den_i, float* den_u,
                               unsigned* nmax_i, unsigned* nmax_u) {
    int i0 = blockIdx.x * blockDim.x + threadIdx.x;
    int st = gridDim.x * blockDim.x;
    for (int i = i0; i < ni * HID; i += st) agg_i[i] = 0.f;
    for (int i = i0; i < nu * HID; i += st) agg_u[i] = 0.f;
    for (int i = i0; i < ni * H; i += st) { den_i[i] = 0.f; nmax_i[i] = 0u; }  // 0 == encode(min)
    for (int i = i0; i < nu * H; i += st) { den_u[i] = 0.f; nmax_u[i] = 0u; }
}

__device__ __forceinline__ unsigned enc_f32(float f) {
    unsigned u = __float_as_uint(f);
    return (u & 0x80000000u) ? ~u : (u | 0x80000000u);
}
__device__ __forceinline__ float dec_f32(unsigned e) {
    unsigned u = (e & 0x80000000u) ? (e & 0x7FFFFFFFu) : ~e;
    return __uint_as_float(u);
}

// Pass A: alpha = <q[dst,h], kr[src,h]> * p[h] / sqrt(DH); atomic-max per (dst,h)
__global__ void edge_alpha_kernel(const float* __restrict__ q, const float* __restrict__ kr,
                                  const int* __restrict__ src, const int* __restrict__ dst,
                                  const float* __restrict__ p, int E,
                                  float* __restrict__ alpha, unsigned* __restrict__ nmax) {
    int t = blockIdx.x * blockDim.x + threadIdx.x;
    if (t >= E * H) return;
    int e = t >> 3, h = t & 7;
    int s = src[e], d = dst[e];
    const float4* qv = (const float4*)(q + (size_t)d * HID + h * DH);
    const float4* kv = (const float4*)(kr + (size_t)s * HID + h * DH);
    float acc = 0.f;
#pragma unroll
    for (int i = 0; i < 4; ++i) {
        float4 a = qv[i], b = kv[i];
        acc += a.x * b.x + a.y * b.y + a.z * b.z + a.w * b.w;
    }
    acc *= p[h] * 0.25f;  // 1/sqrt(16)
    alpha[t] = acc;
    atomicMax(&nmax[(size_t)d * H + h], enc_f32(acc));
}

// Pass B: ex = exp(alpha - max[dst,h]); den += ex (overwrite alpha with ex)
__global__ void edge_exp_kernel(const int* __restrict__ dst, int E,
                                float* __restrict__ alpha,
                                const unsigned* __restrict__ nmax,
                                float* __restrict__ den) {
    int t = blockIdx.x * blockDim.x + threadIdx.x;
    if (t >= E * H) return;
    int e = t >> 3, h = t & 7;
    int d = dst[e];
    float mx = dec_f32(nmax[(size_t)d * H + h]);
    if (!isfinite(mx)) mx = 0.f;
    float ex = __expf(alpha[t] - mx);
    alpha[t] = ex;
    unsafeAtomicAdd(&den[(size_t)d * H + h], ex);
}

// Pass C: agg[dst,h,:] += vr[src,h,:] * ex/(den+eps)
__global__ void edge_agg_kernel(const float* __restrict__ vr,
                                const int* __restrict__ src, const int* __restrict__ dst, int E,
                                const float* __restrict__ alpha, const float* __restrict__ den,
                                float* __restrict__ agg) {
    int t = blockIdx.x * blockDim.x + threadIdx.x;
    if (t >= E * H) return;
    int e = t >> 3, h = t & 7;
    int s = src[e], d = dst[e];
    float w = alpha[t] / (den[(size_t)d * H + h] + 1e-16f);
    const float4* vv = (const float4*)(vr + (size_t)s * HID + h * DH);
    float* ag = agg + (size_t)d * HID + h * DH;
#pragma unroll
    for (int i = 0; i < 4; ++i) {
        float4 v4 = vv[i];
        unsafeAtomicAdd(&ag[i * 4 + 0], v4.x * w);
        unsafeAtomicAdd(&ag[i * 4 + 1], v4.y * w);
        unsafeAtomicAdd(&ag[i * 4 + 2], v4.z * w);
        unsafeAtomicAdd(&ag[i * 4 + 3], v4.w * w);
    }
}

// ---------------------------------------------------------------------------
extern "C" void kernel_launch(void* const* d_in, const int* in_sizes, int n_in,
                              void* d_out, int out_size, void* d_ws, size_t ws_size,
                              hipStream_t stream) {
    const float* x_user = (const float*)d_in[0];
    const float* x_item = (const float*)d_in[1];
    const float* x_taste = (const float*)d_in[2];
    const float* x_image = (const float*)d_in[3];
    const float* Wk = (const float*)d_in[4];
    const float* bk = (const float*)d_in[5];
    const float* Wq = (const float*)d_in[6];
    const float* bq = (const float*)d_in[7];
    const float* Wv = (const float*)d_in[8];
    const float* bv = (const float*)d_in[9];
    const float* a_rel = (const float*)d_in[10];
    const float* m_rel = (const float*)d_in[11];
    const float* p_rel = (const float*)d_in[12];
    const float* Wo = (const float*)d_in[13];
    const float* bo = (const float*)d_in[14];
    const float* skip = (const float*)d_in[15];
    const int* ti_src = (const int*)d_in[16];
    const int* ti_dst = (const int*)d_in[17];
    const int* im_src = (const int*)d_in[18];
    const int* im_dst = (const int*)d_in[19];
    const int* ub_src = (const int*)d_in[20];
    const int* ub_dst = (const int*)d_in[21];
    const int* bu_src = (const int*)d_in[22];
    const int* bu_dst = (const int*)d_in[23];

    const int NU = in_sizes[0] / HID, NI = in_sizes[1] / HID;
    const int NT = in_sizes[2] / HID, NG = in_sizes[3] / HID;
    const int Eti = in_sizes[16], Eim = in_sizes[18];
    const int Eub = in_sizes[20], Ebu = in_sizes[22];

    // workspace layout (floats)
    float* w = (float*)d_ws;
    size_t off = 0;
    auto alloc = [&](size_t n) { float* p = w + off; off += n; return p; };
    float* q_u  = alloc((size_t)NU * HID);
    float* q_i  = alloc((size_t)NI * HID);
    float* kr_u = alloc((size_t)NU * HID);
    float* vr_u = alloc((size_t)NU * HID);
    float* kr_i = alloc((size_t)NI * HID);
    float* vr_i = alloc((size_t)NI * HID);
    float* kr_t = alloc((size_t)NT * HID);
    float* vr_t = alloc((size_t)NT * HID);
    float* kr_g = alloc((size_t)NG * HID);
    float* vr_g = alloc((size_t)NG * HID);
    float* Wka_p = alloc(4 * HID * HID);   // packed
    float* bka   = alloc(4 * HID);
    float* Wva_p = alloc(4 * HID * HID);   // packed
    float* bva   = alloc(4 * HID);
    float* Wq_p  = alloc(2 * HID * HID);   // packed Wq[0..1]
    float* Wo_p  = alloc(2 * HID * HID);   // packed Wo[0..1]
    float* alpha_i = alloc((size_t)(Eti + Eim + Eub) * H);
    float* alpha_u = alloc((size_t)Ebu * H);
    float* den_i = alloc((size_t)NI * H);
    float* den_u = alloc((size_t)NU * H);
    unsigned* nmax_i = (unsigned*)alloc((size_t)NI * H);
    unsigned* nmax_u = (unsigned*)alloc((size_t)NU * H);
    float* agg_i = alloc((size_t)NI * HID);
    float* agg_u = alloc((size_t)NU * HID);

    float* out_i = (float*)d_out;
    float* out_u = (float*)d_out + (size_t)NI * HID;

    // 1) fuse relation transforms into (packed) projection weights; repack Wq/Wo
    fuse_weights_kernel<<<(4 * HID * HID + 255) / 256, 256, 0, stream>>>(
        Wk, bk, Wv, bv, a_rel, m_rel, Wka_p, bka, Wva_p, bva);
    repack_kernel<<<(2 * HID * HID + 255) / 256, 256, 0, stream>>>(Wq, Wq_p, 2);
    repack_kernel<<<(2 * HID * HID + 255) / 256, 256, 0, stream>>>(Wo, Wo_p, 2);

    // 2) zero accumulators
    init_ws_kernel<<<(NI * HID + 255) / 256, 256, 0, stream>>>(
        agg_i, NI, agg_u, NU, den_i, den_u, nmax_i, nmax_u);

    // 3) node projections via WMMA GEMMs (kr/vr use fused weights, q uses Wq)
    auto gemm = [&](const float* X, const float* Wp_, const float* b_, float* O, int N) {
        gemm32_kernel<<<(N + 31) / 32, 32, 0, stream>>>(X, Wp_, b_, O, N);
    };
    gemm(x_user, Wq_p + 0 * HID * HID, bq + 0 * HID, q_u, NU);
    gemm(x_item, Wq_p + 1 * HID * HID, bq + 1 * HID, q_i, NI);
    gemm(x_user, Wka_p + 0 * HID * HID, bka + 0 * HID, kr_u, NU);
    gemm(x_user, Wva_p + 0 * HID * HID, bva + 0 * HID, vr_u, NU);
    gemm(x_item, Wka_p + 1 * HID * HID, bka + 1 * HID, kr_i, NI);
    gemm(x_item, Wva_p + 1 * HID * HID, bva + 1 * HID, vr_i, NI);
    gemm(x_taste, Wka_p + 2 * HID * HID, bka + 2 * HID, kr_t, NT);
    gemm(x_taste, Wva_p + 2 * HID * HID, bva + 2 * HID, vr_t, NT);
    gemm(x_image, Wka_p + 3 * HID * HID, bka + 3 * HID, kr_g, NG);
    gemm(x_image, Wva_p + 3 * HID * HID, bva + 3 * HID, vr_g, NG);

    float* al0 = alpha_i;
    float* al1 = alpha_i + (size_t)Eti * H;
    float* al2 = alpha_i + (size_t)(Eti + Eim) * H;

    // 4) pass A: alpha + segment max
    auto grid8 = [](int E) { return (E * H + 255) / 256; };
    edge_alpha_kernel<<<grid8(Eti), 256, 0, stream>>>(q_i, kr_t, ti_src, ti_dst, p_rel + 0 * H, Eti, al0, nmax_i);
    edge_alpha_kernel<<<grid8(Eim), 256, 0, stream>>>(q_i, kr_g, im_src, im_dst, p_rel + 1 * H, Eim, al1, nmax_i);
    edge_alpha_kernel<<<grid8(Eub), 256, 0, stream>>>(q_i, kr_u, ub_src, ub_dst, p_rel + 2 * H, Eub, al2, nmax_i);
    edge_alpha_kernel<<<grid8(Ebu), 256, 0, stream>>>(q_u, kr_i, bu_src, bu_dst, p_rel + 3 * H, Ebu, alpha_u, nmax_u);

    // 5) pass B: exp + denominator
    edge_exp_kernel<<<grid8(Eti), 256, 0, stream>>>(ti_dst, Eti, al0, nmax_i, den_i);
    edge_exp_kernel<<<grid8(Eim), 256, 0, stream>>>(im_dst, Eim, al1, nmax_i, den_i);
    edge_exp_kernel<<<grid8(Eub), 256, 0, stream>>>(ub_dst, Eub, al2, nmax_i, den_i);
    edge_exp_kernel<<<grid8(Ebu), 256, 0, stream>>>(bu_dst, Ebu, alpha_u, nmax_u, den_u);

    // 6) pass C: weighted message aggregation
    edge_agg_kernel<<<grid8(Eti), 256, 0, stream>>>(vr_t, ti_src, ti_dst, Eti, al0, den_i, agg_i);
    edge_agg_kernel<<<grid8(Eim), 256, 0, stream>>>(vr_g, im_src, im_dst, Eim, al1, den_i, agg_i);
    edge_agg_kernel<<<grid8(Eub), 256, 0, stream>>>(vr_u, ub_src, ub_dst, Eub, al2, den_i, agg_i);
    edge_agg_kernel<<<grid8(Ebu), 256, 0, stream>>>(vr_i, bu_src, bu_dst, Ebu, alpha_u, den_u, agg_u);

    // 7) output: gelu -> GEMM -> gated skip
    out_gemm_kernel<<<(NI + 31) / 32, 32, 0, stream>>>(agg_i, Wo_p + 0 * HID * HID, bo + 0 * HID,
                                                       x_item, skip, 0, out_i, NI);
    out_gemm_kernel<<<(NU + 31) / 32, 32, 0, stream>>>(agg_u, Wo_p + 1 * HID * HID, bo + 1 * HID,
                                                       x_user, skip, 1, out_u, NU);
}